// EdgeEncoding_73804718015011
// MI455X (gfx1250) — compile-verified
//
#include <hip/hip_runtime.h>

typedef __attribute__((ext_vector_type(2))) float v2f;
typedef __attribute__((ext_vector_type(8))) float v8f;
typedef __attribute__((ext_vector_type(4))) float f4;

#define LMAX 5

// ---------------------------------------------------------------------------
// Kernel 1: zero-fill the 256 MB dense output with nontemporal float4 stores
// (don't pollute L2 -- edge_attr wants to stay resident there for kernel 2).
// ---------------------------------------------------------------------------
__global__ void fill_zero_kernel(f4* __restrict__ out, long n4) {
  long i = (long)blockIdx.x * blockDim.x + threadIdx.x;
  if (i < n4) {
    f4 z = {0.f, 0.f, 0.f, 0.f};
    __builtin_nontemporal_store(z, out + i);
  }
}

// ---------------------------------------------------------------------------
// Kernel 2: per-wave WMMA path encoder.
// Each wave32 owns 16 paths. For each position j in 0..4:
//   A (16x64 f32)  = edge_attr rows of the 16 paths' j-th edge
//   B (64x16 f32)  = edge_vector^T (5 real columns, rest zero)
//   C = A @ B via 16 chained V_WMMA_F32_16X16X4_F32 (K=4 each)
// C goes through LDS so row m can pick its data-dependent column pos[m][j].
// ---------------------------------------------------------------------------
__global__ void __launch_bounds__(256)
edge_encoding_wmma_kernel(const float* __restrict__ edge_attr,
                          const float* __restrict__ edge_vector,
                          const int*   __restrict__ path_idx,
                          const int*   __restrict__ src,
                          const int*   __restrict__ dst,
                          float*       __restrict__ out,
                          int E, int P, int N) {
  __shared__ float ldsEV[LMAX * 64];   // edge_vector copy
  __shared__ float ldsC[8 * 256];      // one 16x16 C tile per wave (8 waves)

  const int t = threadIdx.x;
  for (int i = t; i < LMAX * 64; i += blockDim.x) ldsEV[i] = edge_vector[i];
  __syncthreads();

  const int lane = t & 31;
  const int wave = t >> 5;
  const int m    = lane & 15;   // row (path-within-group) this lane serves
  const int half = lane >> 4;   // 0 or 1 -> K sub-offset inside each K=4 group
  const int koff = 2 * half;

  // B fragments: lane l holds B[k][n] with n = l&15, k = 4*kk + koff + {0,1}.
  // B[k][n] = edge_vector[n][k]; columns n >= 5 are zero.
  v2f bfrag[16];
#pragma unroll
  for (int kk = 0; kk < 16; ++kk) {
    float b0 = 0.f, b1 = 0.f;
    if (m < LMAX) {
      b0 = ldsEV[m * 64 + kk * 4 + koff];
      b1 = ldsEV[m * 64 + kk * 4 + koff + 1];
    }
    bfrag[kk][0] = b0;
    bfrag[kk][1] = b1;
  }

  const long group = (long)blockIdx.x * (blockDim.x >> 5) + wave;
  const long base  = group * 16;
  if (base >= P) return;                 // wave-uniform: EXEC stays full
  const long p    = base + m;
  const bool prow = (p < P);

  // Per-row path metadata (lanes l and l+16 mirror row l&15).
  int idxs[LMAX], mask[LMAX], pos[LMAX];
  int run = 0;
#pragma unroll
  for (int j = 0; j < LMAX; ++j) {
    int v   = prow ? path_idx[p * LMAX + j] : -1;
    idxs[j] = v;
    int mk  = (v >= 0 && v < E) ? 1 : 0;
    mask[j] = mk;
    run    += mk;
    pos[j]  = run - 1;                   // compaction position
  }

  float* cTile = &ldsC[wave * 256];
  float  sum = 0.f, cnt = 0.f;

#pragma unroll
  for (int j = 0; j < LMAX; ++j) {
    int e = idxs[j];
    e = e < 0 ? 0 : (e >= E ? E - 1 : e);          // clip like the reference
    const float* aptr = edge_attr + ((size_t)e << 6) + koff;

    v8f c = {};
#pragma unroll
    for (int kk = 0; kk < 16; ++kk) {
      v2f a = *(const v2f*)(aptr + kk * 4);        // 8B aligned (even offset)
      c = __builtin_amdgcn_wmma_f32_16x16x4_f32(
              /*neg_a=*/false, a, /*neg_b=*/false, bfrag[kk],
              /*c_mod=*/(short)0, c, /*reuse_a=*/false, /*reuse_b=*/false);
    }

    // C layout: lane l, vgpr v holds C[v + 8*(l>>4)][l&15].
    __builtin_amdgcn_wave_barrier();
#pragma unroll
    for (int v = 0; v < 8; ++v) {
      int row = v + 8 * half;
      cTile[row * 16 + m] = c[v];
    }
    __builtin_amdgcn_wave_barrier();
    // Same-wave LDS ops are in-order: this load observes the stores above.
    if (mask[j]) {
      int pp = pos[j];
      pp = pp < 0 ? 0 : (pp > 15 ? 15 : pp);
      sum += cTile[m * 16 + pp];
      cnt += 1.f;
    }
    __builtin_amdgcn_wave_barrier();
  }

  if (lane < 16 && prow) {
    float val = cnt > 0.f ? sum / cnt : 0.f;
    long  o   = (long)src[p] * N + dst[p];
    __builtin_nontemporal_store(val, &out[o]);     // (src,dst) pairs unique
  }
}

// ---------------------------------------------------------------------------
extern "C" void kernel_launch(void* const* d_in, const int* in_sizes, int n_in,
                              void* d_out, int out_size, void* d_ws, size_t ws_size,
                              hipStream_t stream) {
  // setup_inputs order: x, edge_attr, edge_vector, path_idx, path_len, src, dst
  const float* edge_attr   = (const float*)d_in[1];
  const float* edge_vector = (const float*)d_in[2];
  const int*   path_idx    = (const int*)d_in[3];
  const int*   src         = (const int*)d_in[5];
  const int*   dst         = (const int*)d_in[6];
  float*       out         = (float*)d_out;

  const int L = LMAX;
  const int D = in_sizes[2] / L;   // 64
  const int N = in_sizes[0] / D;   // 8192
  const int E = in_sizes[1] / D;   // 262144
  const int P = in_sizes[5];       // 524288
  (void)D; (void)n_in; (void)d_ws; (void)ws_size;

  // 1) zero-fill output (out_size = N*N, divisible by 4)
  long n4 = (long)out_size / 4;
  int  threads = 256;
  long fillBlocks = (n4 + threads - 1) / threads;
  fill_zero_kernel<<<(unsigned)fillBlocks, threads, 0, stream>>>((f4*)out, n4);

  // 2) WMMA path encoding + scatter: one wave per 16 paths, 8 waves/block
  long groups = ((long)P + 15) / 16;
  long blocks = (groups + 7) / 8;
  edge_encoding_wmma_kernel<<<(unsigned)blocks, 256, 0, stream>>>(
      edge_attr, edge_vector, path_idx, src, dst, out, E, P, N);
}